// HybridAttention_83923660964345
// MI455X (gfx1250) — compile-verified
//
#include <hip/hip_runtime.h>
#include <math.h>

typedef __attribute__((ext_vector_type(2))) float v2f;
typedef __attribute__((ext_vector_type(8))) float v8f;

#define SEQ    1024
#define DIM_E  448
#define DIM_H  64
#define TOTAL  512

#define WMMA_F32(a, b, c) \
    __builtin_amdgcn_wmma_f32_16x16x4_f32(false, (a), false, (b), (short)0, (c), false, false)

// workspace layout (in floats)
constexpr size_t OFF_Q    = 0;
constexpr size_t OFF_K    = 524288;            // 1024*512
constexpr size_t OFF_V    = 1048576;
constexpr size_t OFF_QH   = 1572864;           // 1024*64
constexpr size_t OFF_KH   = OFF_QH + 65536;
constexpr size_t OFF_QN   = OFF_KH + 65536;
constexpr size_t OFF_KN   = OFF_QN + 1024;
constexpr size_t OFF_X2Q  = OFF_KN + 1024;
constexpr size_t OFF_X2K  = OFF_X2Q + 1024;
constexpr size_t OFF_RSUM = OFF_X2K + 1024;
constexpr size_t OFF_P    = OFF_RSUM + 1024;   // 1024*1024
constexpr size_t OFF_O    = OFF_P + 1048576;

// ---------------------------------------------------------------------------
// Y[m,n] = sum_k A[m,k]*W[n,k] + bias[n]   (A: MxK row-major, W: NxK row-major)
// One wave computes a 16x64 tile (4 WMMA accumulators). K is processed in
// chunks of 16: a load phase (20 b64 loads in flight) then 16 WMMAs, so the
// scheduler can stage s_wait_loadcnt instead of stalling per WMMA.
// ---------------------------------------------------------------------------
__global__ void gemm_wt_kernel(const float* __restrict__ A,
                               const float* __restrict__ W,
                               const float* __restrict__ bias,
                               float* __restrict__ Y,
                               int M, int N, int Kdim)
{
    const int wave = threadIdx.x >> 5;
    const int lane = threadIdx.x & 31;
    const int tile = blockIdx.x * (blockDim.x >> 5) + wave;
    const int ntn  = N >> 6;                     // 64-wide column tiles
    const int tm = tile / ntn;
    const int tn = tile % ntn;
    if (tm >= (M >> 4)) return;

    const int m = lane & 15;
    const int h = lane >> 4;
    const int n0 = tn * 64;

    const float* arow  = A + (size_t)(tm * 16 + m) * Kdim;
    const float* wrow0 = W + (size_t)(n0 +  0 + m) * Kdim;
    const float* wrow1 = W + (size_t)(n0 + 16 + m) * Kdim;
    const float* wrow2 = W + (size_t)(n0 + 32 + m) * Kdim;
    const float* wrow3 = W + (size_t)(n0 + 48 + m) * Kdim;

    v8f acc0 = {}, acc1 = {}, acc2 = {}, acc3 = {};
    for (int kb = 0; kb < Kdim; kb += 16) {
        v2f a[4], b0[4], b1[4], b2[4], b3[4];
        #pragma unroll
        for (int u = 0; u < 4; ++u) {            // load phase
            const int k = kb + 4 * u + 2 * h;
            a [u] = *(const v2f*)(arow  + k);
            b0[u] = *(const v2f*)(wrow0 + k);
            b1[u] = *(const v2f*)(wrow1 + k);
            b2[u] = *(const v2f*)(wrow2 + k);
            b3[u] = *(const v2f*)(wrow3 + k);
        }
        #pragma unroll
        for (int u = 0; u < 4; ++u) {            // compute phase
            acc0 = WMMA_F32(a[u], b0[u], acc0);
            acc1 = WMMA_F32(a[u], b1[u], acc1);
            acc2 = WMMA_F32(a[u], b2[u], acc2);
            acc3 = WMMA_F32(a[u], b3[u], acc3);
        }
    }

    const float bn0 = bias[n0 +  0 + m];
    const float bn1 = bias[n0 + 16 + m];
    const float bn2 = bias[n0 + 32 + m];
    const float bn3 = bias[n0 + 48 + m];
    float* yb = Y + (size_t)(tm * 16 + 8 * h) * N + n0 + m;
    #pragma unroll
    for (int r = 0; r < 8; ++r) {                // D row = r + 8h, col = 16j + m
        yb[(size_t)r * N +  0] = acc0[r] + bn0;
        yb[(size_t)r * N + 16] = acc1[r] + bn1;
        yb[(size_t)r * N + 32] = acc2[r] + bn2;
        yb[(size_t)r * N + 48] = acc3[r] + bn3;
    }
}

// ---------------------------------------------------------------------------
// Per-row stats: qn = |x[:448]|^2 ; exp0 of x[448:512] -> XH ; x2 = |XH|^2
// ---------------------------------------------------------------------------
__global__ void rowstats_kernel(const float* __restrict__ X,
                                float* __restrict__ rn,
                                float* __restrict__ XH,
                                float* __restrict__ x2out)
{
    const int lane = threadIdx.x & 31;
    const int row  = blockIdx.x * (blockDim.x >> 5) + (threadIdx.x >> 5);
    if (row >= SEQ) return;
    const float* xr = X + (size_t)row * TOTAL;

    float se = 0.f;
    #pragma unroll
    for (int i = 0; i < DIM_E / 32; ++i) {
        float v = xr[lane + 32 * i];
        se += v * v;
    }
    const float v0 = xr[DIM_E + lane];
    const float v1 = xr[DIM_E + 32 + lane];
    float sh = v0 * v0 + v1 * v1;

    #pragma unroll
    for (int off = 16; off >= 1; off >>= 1) {
        se += __shfl_xor(se, off, 32);
        sh += __shfl_xor(sh, off, 32);
    }

    const float norm  = fmaxf(sqrtf(sh), 1e-5f);
    const float t     = tanhf(norm);
    const float scale = fminf(0.999f / fmaxf(t, 1e-5f), 1.f);
    const float ff    = (t / norm) * scale;

    XH[(size_t)row * DIM_H + lane]      = v0 * ff;
    XH[(size_t)row * DIM_H + 32 + lane] = v1 * ff;
    if (lane == 0) {
        rn[row] = se;
        const float tc = fminf(t, 0.999f);
        x2out[row] = tc * tc;
    }
}

// ---------------------------------------------------------------------------
// Score tile (16 rows x 64 cols per wave):
// P[s,t] = exp(clip(-(d_e + alpha*d_h^2)/sqrt(512), -50, 0))
// ---------------------------------------------------------------------------
__global__ void scores_kernel(const float* __restrict__ Q,
                              const float* __restrict__ Km,
                              const float* __restrict__ QH,
                              const float* __restrict__ KH,
                              const float* __restrict__ qn,
                              const float* __restrict__ kn,
                              const float* __restrict__ x2q,
                              const float* __restrict__ x2k,
                              const float* __restrict__ alpha_un,
                              float* __restrict__ P)
{
    const int wave = threadIdx.x >> 5;
    const int lane = threadIdx.x & 31;
    const int tile = blockIdx.x * (blockDim.x >> 5) + wave;
    const int ntn  = SEQ >> 6;
    const int tm = tile / ntn;
    const int tn = tile % ntn;
    if (tm >= (SEQ >> 4)) return;

    const int m = lane & 15;
    const int h = lane >> 4;
    const int t0 = tn * 64;

    // Euclidean dot products over first 448 dims
    const float* qrow = Q  + (size_t)(tm * 16 + m) * TOTAL;
    const float* kr0  = Km + (size_t)(t0 +  0 + m) * TOTAL;
    const float* kr1  = Km + (size_t)(t0 + 16 + m) * TOTAL;
    const float* kr2  = Km + (size_t)(t0 + 32 + m) * TOTAL;
    const float* kr3  = Km + (size_t)(t0 + 48 + m) * TOTAL;

    v8f ae0 = {}, ae1 = {}, ae2 = {}, ae3 = {};
    for (int kb = 0; kb < DIM_E; kb += 16) {
        v2f a[4], b0[4], b1[4], b2[4], b3[4];
        #pragma unroll
        for (int u = 0; u < 4; ++u) {
            const int k = kb + 4 * u + 2 * h;
            a [u] = *(const v2f*)(qrow + k);
            b0[u] = *(const v2f*)(kr0  + k);
            b1[u] = *(const v2f*)(kr1  + k);
            b2[u] = *(const v2f*)(kr2  + k);
            b3[u] = *(const v2f*)(kr3  + k);
        }
        #pragma unroll
        for (int u = 0; u < 4; ++u) {
            ae0 = WMMA_F32(a[u], b0[u], ae0);
            ae1 = WMMA_F32(a[u], b1[u], ae1);
            ae2 = WMMA_F32(a[u], b2[u], ae2);
            ae3 = WMMA_F32(a[u], b3[u], ae3);
        }
    }

    // Hyperbolic dot products over mapped 64 dims
    const float* qh  = QH + (size_t)(tm * 16 + m) * DIM_H;
    const float* kh0 = KH + (size_t)(t0 +  0 + m) * DIM_H;
    const float* kh1 = KH + (size_t)(t0 + 16 + m) * DIM_H;
    const float* kh2 = KH + (size_t)(t0 + 32 + m) * DIM_H;
    const float* kh3 = KH + (size_t)(t0 + 48 + m) * DIM_H;

    v8f ah0 = {}, ah1 = {}, ah2 = {}, ah3 = {};
    for (int kb = 0; kb < DIM_H; kb += 16) {
        v2f a[4], b0[4], b1[4], b2[4], b3[4];
        #pragma unroll
        for (int u = 0; u < 4; ++u) {
            const int k = kb + 4 * u + 2 * h;
            a [u] = *(const v2f*)(qh  + k);
            b0[u] = *(const v2f*)(kh0 + k);
            b1[u] = *(const v2f*)(kh1 + k);
            b2[u] = *(const v2f*)(kh2 + k);
            b3[u] = *(const v2f*)(kh3 + k);
        }
        #pragma unroll
        for (int u = 0; u < 4; ++u) {
            ah0 = WMMA_F32(a[u], b0[u], ah0);
            ah1 = WMMA_F32(a[u], b1[u], ah1);
            ah2 = WMMA_F32(a[u], b2[u], ah2);
            ah3 = WMMA_F32(a[u], b3[u], ah3);
        }
    }

    const float au    = alpha_un[0];
    const float alpha = (au > 20.f) ? au : log1pf(expf(au));  // softplus
    const float inv_s = 0.044194173824159216f;                // 1/sqrt(512)

    float knv[4], y2v[4];
    #pragma unroll
    for (int j = 0; j < 4; ++j) {
        knv[j] = kn [t0 + 16 * j + m];
        y2v[j] = x2k[t0 + 16 * j + m];
    }

    #pragma unroll
    for (int r = 0; r < 8; ++r) {
        const int   srow = tm * 16 + r + 8 * h;
        const float qnv  = qn [srow];
        const float x2   = x2q[srow];
        float ev[4] = { ae0[r], ae1[r], ae2[r], ae3[r] };
        float hv[4] = { ah0[r], ah1[r], ah2[r], ah3[r] };
        #pragma unroll
        for (int j = 0; j < 4; ++j) {
            const float de = qnv + knv[j] - 2.f * ev[j];
            const float xy = hv[j];
            const float y2 = y2v[j];
            // mobius_add(-x, y, c=1): num = -Af*x + Bf*y ; den = 1 - 2xy + x2*y2
            const float Af  = 1.f - 2.f * xy + y2;
            const float Bf  = 1.f - x2;
            const float n2  = Af * Af * x2 + Bf * Bf * y2 - 2.f * Af * Bf * xy;
            const float den = fmaxf(1.f - 2.f * xy + x2 * y2, 1e-5f);
            float nv = sqrtf(fmaxf(n2, 0.f)) / den;
            nv = fminf(nv, 0.999f);                           // project + dist clamps
            const float dh   = 2.f * atanhf(nv);
            const float dist = de + alpha * dh * dh;
            const float lg   = fminf(fmaxf(-dist * inv_s, -50.f), 0.f);
            P[(size_t)srow * SEQ + t0 + 16 * j + m] = expf(lg);
        }
    }
}

// one wave per row: rsum[s] = sum_t P[s,t]
__global__ void rowsum_kernel(const float* __restrict__ P, float* __restrict__ rsum)
{
    const int lane = threadIdx.x & 31;
    const int row  = blockIdx.x * (blockDim.x >> 5) + (threadIdx.x >> 5);
    if (row >= SEQ) return;
    const float* pr = P + (size_t)row * SEQ;
    float s = 0.f;
    #pragma unroll
    for (int i = 0; i < SEQ / 32; ++i) s += pr[lane + 32 * i];
    #pragma unroll
    for (int off = 16; off >= 1; off >>= 1) s += __shfl_xor(s, off, 32);
    if (lane == 0) rsum[row] = s;
}

// O = (P @ V) / rsum[row]   (V row-major KxN, so B[k,n]=V[k*N+n])
__global__ void gemm_pv_kernel(const float* __restrict__ P,
                               const float* __restrict__ V,
                               const float* __restrict__ rsum,
                               float* __restrict__ O,
                               int M, int N, int Kdim)
{
    const int wave = threadIdx.x >> 5;
    const int lane = threadIdx.x & 31;
    const int tile = blockIdx.x * (blockDim.x >> 5) + wave;
    const int ntn  = N >> 6;
    const int tm = tile / ntn;
    const int tn = tile % ntn;
    if (tm >= (M >> 4)) return;

    const int m = lane & 15;
    const int h = lane >> 4;
    const int n0 = tn * 64;

    const float* prow = P + (size_t)(tm * 16 + m) * Kdim;
    v8f acc0 = {}, acc1 = {}, acc2 = {}, acc3 = {};
    for (int kb = 0; kb < Kdim; kb += 16) {
        v2f a[4], b0[4], b1[4], b2[4], b3[4];
        #pragma unroll
        for (int u = 0; u < 4; ++u) {
            const int k0 = kb + 4 * u + 2 * h;
            const float* v0p = V + (size_t)k0 * N + n0 + m;
            const float* v1p = v0p + N;
            a [u] = *(const v2f*)(prow + k0);
            b0[u] = (v2f){ v0p[ 0], v1p[ 0] };
            b1[u] = (v2f){ v0p[16], v1p[16] };
            b2[u] = (v2f){ v0p[32], v1p[32] };
            b3[u] = (v2f){ v0p[48], v1p[48] };
        }
        #pragma unroll
        for (int u = 0; u < 4; ++u) {
            acc0 = WMMA_F32(a[u], b0[u], acc0);
            acc1 = WMMA_F32(a[u], b1[u], acc1);
            acc2 = WMMA_F32(a[u], b2[u], acc2);
            acc3 = WMMA_F32(a[u], b3[u], acc3);
        }
    }
    #pragma unroll
    for (int r = 0; r < 8; ++r) {
        const int row = tm * 16 + r + 8 * h;
        const float inv = 1.f / rsum[row];
        float* ob = O + (size_t)row * N + n0 + m;
        ob[ 0] = acc0[r] * inv;
        ob[16] = acc1[r] * inv;
        ob[32] = acc2[r] * inv;
        ob[48] = acc3[r] * inv;
    }
}

// ---------------------------------------------------------------------------
extern "C" void kernel_launch(void* const* d_in, const int* in_sizes, int n_in,
                              void* d_out, int out_size, void* d_ws, size_t ws_size,
                              hipStream_t stream)
{
    const float* x   = (const float*)d_in[0];
    const float* wq  = (const float*)d_in[1];
    const float* bq  = (const float*)d_in[2];
    const float* wk  = (const float*)d_in[3];
    const float* bk  = (const float*)d_in[4];
    const float* wv  = (const float*)d_in[5];
    const float* bv  = (const float*)d_in[6];
    const float* wo  = (const float*)d_in[7];
    const float* bo  = (const float*)d_in[8];
    const float* alp = (const float*)d_in[9];

    float* ws   = (float*)d_ws;
    float* Q    = ws + OFF_Q;
    float* Kt   = ws + OFF_K;
    float* V    = ws + OFF_V;
    float* QH   = ws + OFF_QH;
    float* KH   = ws + OFF_KH;
    float* qn   = ws + OFF_QN;
    float* kn   = ws + OFF_KN;
    float* x2q  = ws + OFF_X2Q;
    float* x2k  = ws + OFF_X2K;
    float* rsum = ws + OFF_RSUM;
    float* P    = ws + OFF_P;
    float* O    = ws + OFF_O;

    // (1024/16)*(512/64) = 512 tiles, 4 waves per 128-thread block -> 128 blocks
    gemm_wt_kernel<<<128, 128, 0, stream>>>(x, wq, bq, Q,  SEQ, TOTAL, TOTAL);
    gemm_wt_kernel<<<128, 128, 0, stream>>>(x, wk, bk, Kt, SEQ, TOTAL, TOTAL);
    gemm_wt_kernel<<<128, 128, 0, stream>>>(x, wv, bv, V,  SEQ, TOTAL, TOTAL);

    rowstats_kernel<<<128, 256, 0, stream>>>(Q,  qn, QH, x2q);
    rowstats_kernel<<<128, 256, 0, stream>>>(Kt, kn, KH, x2k);

    // (1024/16)*(1024/64) = 1024 tiles -> 256 blocks
    scores_kernel<<<256, 128, 0, stream>>>(Q, Kt, QH, KH, qn, kn, x2q, x2k, alp, P);

    rowsum_kernel<<<128, 256, 0, stream>>>(P, rsum);

    // (1024/16)*(512/64) = 512 tiles -> 128 blocks, K=1024
    gemm_pv_kernel<<<128, 128, 0, stream>>>(P, V, rsum, O, SEQ, TOTAL, SEQ);

    // final projection straight into d_out
    gemm_wt_kernel<<<128, 128, 0, stream>>>(O, wo, bo, (float*)d_out, SEQ, TOTAL, TOTAL);
}